// ClusterMoCoKnnBert_29764123361922
// MI455X (gfx1250) — compile-verified
//
#include <hip/hip_runtime.h>
#include <hip/hip_bf16.h>

typedef float v2f __attribute__((ext_vector_type(2)));
typedef float v8f __attribute__((ext_vector_type(8)));

#define BQ 32
#define KQ 131072
#define HD 768
#define NEG_INF (-__builtin_inff())

// ---------------------------------------------------------------------------
// GEMM: cos[32][131072] = liner_q[32][768] * feature_queue[131072][768]^T
// One wave computes a 32(M) x 16(N) tile via two V_WMMA_F32_16X16X4_F32
// accumulators sharing the B operand. Epilogue also writes the neg-masked
// copy (pos positions -> -inf) so no second pass over cos_sim is needed.
// ---------------------------------------------------------------------------
__global__ __launch_bounds__(256) void moco_gemm_wmma(
    const float* __restrict__ q, const float* __restrict__ fq,
    const int* __restrict__ label_q, const int* __restrict__ cluster_q,
    const int* __restrict__ label_queue, const int* __restrict__ cluster_queue,
    float* __restrict__ cosv, float* __restrict__ neg) {
  const int lane = threadIdx.x & 31;
  const int wid  = threadIdx.x >> 5;
  const int gw   = blockIdx.x * 8 + wid;       // 8192 waves total
  const int nbase = gw * 16;
  const int l15  = lane & 15;
  const int hoff = (lane >> 4) * 2;            // 0 for lanes 0-15, 2 for 16-31

  const float* bptr  = fq + (size_t)(nbase + l15) * HD + hoff;
  const float* aptr0 = q  + (size_t)l15 * HD + hoff;
  const float* aptr1 = q  + (size_t)(l15 + 16) * HD + hoff;

  v8f c0 = {};
  v8f c1 = {};
#pragma unroll 4
  for (int h = 0; h < HD; h += 4) {
    v2f a0 = *(const v2f*)(aptr0 + h);
    v2f a1 = *(const v2f*)(aptr1 + h);
    v2f b  = *(const v2f*)(bptr + h);
    c0 = __builtin_amdgcn_wmma_f32_16x16x4_f32(false, a0, false, b, (short)0,
                                               c0, false, false);
    c1 = __builtin_amdgcn_wmma_f32_16x16x4_f32(false, a1, false, b, (short)0,
                                               c1, false, false);
  }

  // C/D layout: vgpr r, lanes 0-15 -> (M=r, N=lane); lanes 16-31 -> (M=r+8, N=lane-16)
  const int ncol = nbase + l15;
  const int mhi  = (lane >> 4) * 8;
  const int cqv  = cluster_queue[ncol];
  const int lqv  = label_queue[ncol];
#pragma unroll
  for (int r = 0; r < 8; ++r) {
    const int m0 = r + mhi;
    const int m1 = m0 + 16;
    const float v0 = c0[r];
    const float v1 = c1[r];
    cosv[(size_t)m0 * KQ + ncol] = v0;
    cosv[(size_t)m1 * KQ + ncol] = v1;
    const bool p0 = ((cqv == cluster_q[m0]) == (lqv == label_q[m0]));
    const bool p1 = ((cqv == cluster_q[m1]) == (lqv == label_q[m1]));
    neg[(size_t)m0 * KQ + ncol] = p0 ? NEG_INF : v0;
    neg[(size_t)m1 * KQ + ncol] = p1 ? NEG_INF : v1;
  }
}

// ---------------------------------------------------------------------------
// Per-row pos counts from the int arrays only (no cos_sim traffic).
// Wave ballot + popcount, LDS partials, one global atomic per row per block.
// ---------------------------------------------------------------------------
__global__ __launch_bounds__(256) void moco_count(
    const int* __restrict__ label_q, const int* __restrict__ cluster_q,
    const int* __restrict__ label_queue, const int* __restrict__ cluster_queue,
    int* __restrict__ posCnt) {
  __shared__ int slq[BQ], scq[BQ], cnt[BQ];
  const int tid = threadIdx.x;
  if (tid < BQ) {
    slq[tid] = label_q[tid];
    scq[tid] = cluster_q[tid];
    cnt[tid] = 0;
  }
  __syncthreads();
  const int col = blockIdx.x * 256 + tid;
  const int cqv = cluster_queue[col];
  const int lqv = label_queue[col];
  const int lane = tid & 31;
#pragma unroll 4
  for (int row = 0; row < BQ; ++row) {
    const bool pos = ((cqv == scq[row]) == (lqv == slq[row]));
    const unsigned long long bal = __ballot(pos);
    if (lane == 0) atomicAdd(&cnt[row], (int)__popcll(bal));
  }
  __syncthreads();
  if (tid < BQ && cnt[tid] != 0) atomicAdd(&posCnt[tid], cnt[tid]);
}

// ---------------------------------------------------------------------------
// pos_min = min(min_row posCnt, top_k); neg_min = K - max_row posCnt
// ---------------------------------------------------------------------------
__global__ void moco_mins(const int* __restrict__ posCnt,
                          const int* __restrict__ topk, int* __restrict__ mins) {
  if (threadIdx.x == 0) {
    int mn = 0x7fffffff, mx = -1;
    for (int r = 0; r < BQ; ++r) {
      int c = posCnt[r];
      mn = (c < mn) ? c : mn;
      mx = (c > mx) ? c : mx;
    }
    int tk = topk[0];
    int pos_min = (mn < tk) ? mn : tk;
    if (pos_min < 1) pos_min = 1;            // guard (reference guarantees > 0)
    if (pos_min > 10) pos_min = 10;
    int neg_min = KQ - mx;
    if (neg_min < 1) neg_min = 1;
    mins[0] = pos_min;
    mins[1] = neg_min;
  }
}

// ---------------------------------------------------------------------------
// Per-row top-10 of pos-masked values. Register top-10 per thread, then a
// descending LDS bitonic sort of 4096 candidates; first 10 are the answer.
// ---------------------------------------------------------------------------
__global__ __launch_bounds__(256) void moco_postop(
    const float* __restrict__ cosv, const int* __restrict__ label_q,
    const int* __restrict__ cluster_q, const int* __restrict__ label_queue,
    const int* __restrict__ cluster_queue, float* __restrict__ posTop) {
  __shared__ float s[4096];
  const int row = blockIdx.x;
  const int tid = threadIdx.x;
  const int lq = label_q[row];
  const int cq = cluster_q[row];

  float t[10];
#pragma unroll
  for (int i = 0; i < 10; ++i) t[i] = NEG_INF;

  for (int col = tid; col < KQ; col += 256) {
    const bool pos = ((cluster_queue[col] == cq) == (label_queue[col] == lq));
    const float v = pos ? cosv[(size_t)row * KQ + col] : NEG_INF;
    if (v > t[9]) {
      t[9] = v;
#pragma unroll
      for (int i = 9; i > 0; --i) {
        if (t[i] > t[i - 1]) { float tmp = t[i - 1]; t[i - 1] = t[i]; t[i] = tmp; }
      }
    }
  }
#pragma unroll
  for (int i = 0; i < 10; ++i) s[tid * 10 + i] = t[i];
  for (int i = 2560 + tid; i < 4096; i += 256) s[i] = NEG_INF;
  __syncthreads();

  for (int k = 2; k <= 4096; k <<= 1) {
    for (int j = k >> 1; j > 0; j >>= 1) {
      for (int pair = tid; pair < 2048; pair += 256) {
        const int i = 2 * pair - (pair & (j - 1));
        const int ixj = i | j;
        const float a = s[i];
        const float b = s[ixj];
        const bool desc = ((i & k) == 0);
        if (desc ? (a < b) : (a > b)) { s[i] = b; s[ixj] = a; }
      }
      __syncthreads();
    }
  }
  if (tid < 10) posTop[row * 10 + tid] = s[tid];
}

// ---------------------------------------------------------------------------
// Phase 1 of the per-row descending sort: fully bitonic-sort each 8192-element
// chunk (32 KB) in LDS. Covers stages k=2..8192 of the global network; the
// chunk's direction alternates per the global (i & k) rule.
// ---------------------------------------------------------------------------
__global__ __launch_bounds__(256) void moco_sort_chunk(float* __restrict__ data) {
  __shared__ float s[8192];
  const int tid = threadIdx.x;
  const int row = blockIdx.x >> 4;          // 16 chunks per row
  const int chunk = blockIdx.x & 15;
  const int gbase = chunk * 8192;           // position within row
  float4* s4 = (float4*)s;
  float4* d4 = (float4*)(data + (size_t)row * KQ + (size_t)gbase);

  for (int i = tid; i < 2048; i += 256) s4[i] = d4[i];
  __syncthreads();

  for (int k = 2; k <= 8192; k <<= 1) {
    for (int j = k >> 1; j > 0; j >>= 1) {
      for (int pair = tid; pair < 4096; pair += 256) {
        const int i = 2 * pair - (pair & (j - 1));
        const int ixj = i | j;
        const float a = s[i];
        const float b = s[ixj];
        const bool desc = (((gbase + i) & k) == 0);
        if (desc ? (a < b) : (a > b)) { s[i] = b; s[ixj] = a; }
      }
      __syncthreads();
    }
  }
  for (int i = tid; i < 2048; i += 256) d4[i] = s4[i];
}

// ---------------------------------------------------------------------------
// One global (k,j) pass, j >= 8192 (partner crosses LDS chunks).
// One thread per compare pair: 32 * 65536 threads.
// ---------------------------------------------------------------------------
__global__ __launch_bounds__(256) void moco_bitonic(float* __restrict__ data,
                                                    int k, int j) {
  const int t = blockIdx.x * 256 + threadIdx.x;   // [0, 32*65536)
  const int row = t >> 16;
  const int pair = t & 65535;
  const int i = 2 * pair - (pair & (j - 1));
  const int ixj = i | j;
  const size_t base = (size_t)row * KQ;
  const float a = data[base + i];
  const float b = data[base + ixj];
  const bool desc = ((i & k) == 0);
  if (desc ? (a < b) : (a > b)) {
    data[base + i] = b;
    data[base + ixj] = a;
  }
}

// ---------------------------------------------------------------------------
// Tail of a merge stage: passes j=4096..1 all stay within an 8192 chunk, so
// run them fused in LDS. For k > 8192 the direction (i & k) is constant per
// chunk.
// ---------------------------------------------------------------------------
__global__ __launch_bounds__(256) void moco_merge_chunk(float* __restrict__ data,
                                                        int k) {
  __shared__ float s[8192];
  const int tid = threadIdx.x;
  const int row = blockIdx.x >> 4;
  const int chunk = blockIdx.x & 15;
  const bool desc = (((chunk * 8192) & k) == 0);
  float4* s4 = (float4*)s;
  float4* d4 = (float4*)(data + (size_t)row * KQ + (size_t)chunk * 8192);

  for (int i = tid; i < 2048; i += 256) s4[i] = d4[i];
  __syncthreads();

  for (int j = 4096; j > 0; j >>= 1) {
    for (int pair = tid; pair < 4096; pair += 256) {
      const int i = 2 * pair - (pair & (j - 1));
      const int ixj = i | j;
      const float a = s[i];
      const float b = s[ixj];
      if (desc ? (a < b) : (a > b)) { s[i] = b; s[ixj] = a; }
    }
    __syncthreads();
  }
  for (int i = tid; i < 2048; i += 256) d4[i] = s4[i];
}

// ---------------------------------------------------------------------------
// logits_con assembly: reproduces torch's
//   cat([pos.view(-1,1), neg.view(-1,1).repeat(1,pos_min).view(-1,neg_min)],-1)/T
// ---------------------------------------------------------------------------
__global__ __launch_bounds__(256) void moco_assemble(
    const float* __restrict__ posTop, const float* __restrict__ neg,
    const int* __restrict__ mins, float* __restrict__ out, long long out_size) {
  const long long t = (long long)blockIdx.x * 256 + threadIdx.x;
  if (t >= out_size) return;
  const int pos_min = mins[0];
  const int neg_min = mins[1];
  const long long cols = (long long)neg_min + 1;
  const long long i = t / cols;
  const long long jj = t - i * cols;
  float v;
  if (jj == 0) {
    const int b = (int)(i / pos_min);
    const int p = (int)(i % pos_min);
    v = posTop[b * 10 + p];
  } else {
    // flattened index through the [B*neg_min, pos_min] repeat buffer
    const long long f = i * (long long)neg_min + (jj - 1);
    const long long denom = (long long)neg_min * (long long)pos_min;
    const int b = (int)(f / denom);
    const int n = (int)((f % denom) / pos_min);
    v = neg[(size_t)b * KQ + n];
  }
  out[t] = v / 0.07f;
}

extern "C" void kernel_launch(void* const* d_in, const int* in_sizes, int n_in,
                              void* d_out, int out_size, void* d_ws, size_t ws_size,
                              hipStream_t stream) {
  const float* liner_q       = (const float*)d_in[0];
  const float* feature_queue = (const float*)d_in[1];
  const int*   label_q       = (const int*)d_in[2];
  const int*   cluster_q     = (const int*)d_in[3];
  const int*   label_queue   = (const int*)d_in[4];
  const int*   cluster_queue = (const int*)d_in[5];
  const int*   topk          = (const int*)d_in[6];

  float* cosv   = (float*)d_ws;                      // 32*131072 floats (16 MB)
  float* neg    = cosv + (size_t)BQ * KQ;            // 16 MB
  float* posTop = neg + (size_t)BQ * KQ;             // 320 floats
  int*   posCnt = (int*)(posTop + BQ * 10);          // 32 ints
  int*   mins   = posCnt + BQ;                       // 2 ints

  hipMemsetAsync(posCnt, 0, BQ * sizeof(int), stream);

  // 1) WMMA GEMM with fused neg-masked epilogue: 8192 N-tiles, 8 waves/block
  moco_gemm_wmma<<<1024, 256, 0, stream>>>(liner_q, feature_queue, label_q,
                                           cluster_q, label_queue, cluster_queue,
                                           cosv, neg);

  // 2) per-row pos counts (int arrays only)
  moco_count<<<KQ / 256, 256, 0, stream>>>(label_q, cluster_q, label_queue,
                                           cluster_queue, posCnt);

  // 3) pos_min / neg_min
  moco_mins<<<1, 32, 0, stream>>>(posCnt, topk, mins);

  // 4) per-row pos top-10
  moco_postop<<<BQ, 256, 0, stream>>>(cosv, label_q, cluster_q, label_queue,
                                      cluster_queue, posTop);

  // 5) per-row descending sort of neg: LDS chunk sort, then merge stages with
  //    only the chunk-crossing passes (j >= 8192) in global memory.
  const int chunkBlocks = BQ * (KQ / 8192);          // 512 blocks
  const int pairBlocks  = (BQ * (KQ / 2)) / 256;     // 8192 blocks
  moco_sort_chunk<<<chunkBlocks, 256, 0, stream>>>(neg);
  for (int k = 16384; k <= KQ; k <<= 1) {
    for (int j = k >> 1; j >= 8192; j >>= 1)
      moco_bitonic<<<pairBlocks, 256, 0, stream>>>(neg, k, j);
    moco_merge_chunk<<<chunkBlocks, 256, 0, stream>>>(neg, k);
  }

  // 6) assemble logits
  const long long os = (long long)out_size;
  const unsigned gridAsm = (unsigned)((os + 255) / 256);
  moco_assemble<<<gridAsm, 256, 0, stream>>>(posTop, neg, mins, (float*)d_out, os);
}